// LinkPred_60103772340328
// MI455X (gfx1250) — compile-verified
//
#include <hip/hip_runtime.h>

typedef __attribute__((ext_vector_type(2))) float v2f;
typedef __attribute__((ext_vector_type(8))) float v8f;

#define N_NODES 50000
#define N_EDGES 640000
#define IN_DIM 128
#define HID 128
#define OUT_DIM 64

// ---------------------------------------------------------------- utilities
__global__ void zero_f32(float* __restrict__ p, int n) {
    int i = blockIdx.x * blockDim.x + threadIdx.x;
    if (i < n) p[i] = 0.0f;
}

__global__ void degree_kernel(const int* __restrict__ dst,
                              float* __restrict__ deg, int n_edges) {
    int e = blockIdx.x * blockDim.x + threadIdx.x;
    if (e < n_edges) atomicAdd(&deg[dst[e]], 1.0f);
}

// One wave (32 lanes) per edge: float4 gather of 128-f32 source row,
// 4 global_atomic_add_f32 per lane into msg[dst]. msg (25.6MB) lives in L2.
__global__ void scatter_add128(const float* __restrict__ feat,
                               const int* __restrict__ src,
                               const int* __restrict__ dst,
                               float* __restrict__ msg, int n_edges) {
    int e = blockIdx.x * (blockDim.x >> 5) + (threadIdx.x >> 5);
    if (e >= n_edges) return;
    int lane = threadIdx.x & 31;
    int s = src[e];
    int d = dst[e];
    const float4 v = *(const float4*)(feat + (size_t)s * 128 + lane * 4);
    float* out = msg + (size_t)d * 128 + lane * 4;
    atomicAdd(out + 0, v.x);
    atomicAdd(out + 1, v.y);
    atomicAdd(out + 2, v.z);
    atomicAdd(out + 3, v.w);
}

// ------------------------------------------------- fused SAGE GEMM via WMMA
// One wave computes a 16(node) x 16(outchan) tile of
//   D = A_self * Wself^T + (msg/deg) * Wneigh^T + bias  [optional ReLU]
// using V_WMMA_F32_16X16X4_F32, K looped in steps of 4.
//
// A (16x4 f32) layout: lane L -> row M=L&15, V0=K(2*hi), V1=K(2*hi+1), hi=L>>4
// B (4x16 f32) layout: lane L -> col N=L&15, V0=K(2*hi), V1=K(2*hi+1)
// C/D (16x16 f32):     lane L, VGPR v -> (M = v + 8*hi, N = L&15)
template <int NCOLS, bool RELU>
__global__ void sage_gemm_wmma(const float* __restrict__ xin,     // [N,128]
                               const float* __restrict__ msg,     // [N,128]
                               const float* __restrict__ deg,     // [N]
                               const float* __restrict__ Wself,   // [NCOLS,128]
                               const float* __restrict__ Wneigh,  // [NCOLS,128]
                               const float* __restrict__ bias,    // [NCOLS]
                               float* __restrict__ out) {         // [N,NCOLS]
    const int K = 128;
    const int wave = threadIdx.x >> 5;   // column tile within block
    const int lane = threadIdx.x & 31;
    const int l15  = lane & 15;
    const int hi   = lane >> 4;          // 0 or 1 -> selects K pair
    const int row0 = blockIdx.x * 16;    // node tile (exactly covers 50000)
    const int col0 = wave * 16;

    const int arow = row0 + l15;                 // A-row this lane serves
    // deg is an integer-valued count: single v_rcp_f32 (1 ulp) instead of
    // the full IEEE divide sequence. Mean aggregation tolerates this.
    const float inv = __builtin_amdgcn_rcpf(fmaxf(deg[arow], 1.0f));

    const float* xrow = xin + (size_t)arow * K;
    const float* mrow = msg + (size_t)arow * K;
    const float* wsc  = Wself  + (size_t)(col0 + l15) * K;  // B col -> W row
    const float* wnc  = Wneigh + (size_t)(col0 + l15) * K;

    v8f acc = {};
#pragma unroll
    for (int k = 0; k < K; k += 4) {
        const int ka = k + 2 * hi;
        v2f a_self = *(const v2f*)(xrow + ka);
        v2f a_nei  = *(const v2f*)(mrow + ka);
        a_nei *= inv;                                   // fold mean-divide
        v2f b_self = *(const v2f*)(wsc + ka);
        v2f b_nei  = *(const v2f*)(wnc + ka);
        acc = __builtin_amdgcn_wmma_f32_16x16x4_f32(
            false, a_self, false, b_self, (short)0, acc, false, false);
        acc = __builtin_amdgcn_wmma_f32_16x16x4_f32(
            false, a_nei, false, b_nei, (short)0, acc, false, false);
    }

    const float bv = bias[col0 + l15];
#pragma unroll
    for (int v = 0; v < 8; ++v) {
        float val = acc[v] + bv;
        if (RELU) val = fmaxf(val, 0.0f);
        out[(size_t)(row0 + v + 8 * hi) * NCOLS + col0 + l15] = val;
    }
}

// ------------------------------------------------------------------ driver
extern "C" void kernel_launch(void* const* d_in, const int* in_sizes, int n_in,
                              void* d_out, int out_size, void* d_ws, size_t ws_size,
                              hipStream_t stream) {
    const float* x       = (const float*)d_in[0];
    const float* W_self1 = (const float*)d_in[1];
    const float* W_neigh1= (const float*)d_in[2];
    const float* b1      = (const float*)d_in[3];
    const float* W_self2 = (const float*)d_in[4];
    const float* W_neigh2= (const float*)d_in[5];
    const float* b2      = (const float*)d_in[6];
    const int*   src     = (const int*)d_in[7];
    const int*   dst     = (const int*)d_in[8];
    float* outp = (float*)d_out;

    float* wsf = (float*)d_ws;
    float* deg = wsf;                          // 50000 (padded to 50048)
    float* msg = wsf + 50048;                  // 50000*128
    float* h1  = msg + (size_t)N_NODES * HID;  // 50000*128

    const int ZB = 256;
    // zero deg + msg in one shot (contiguous)
    {
        int n = 50048 + N_NODES * HID;
        zero_f32<<<(n + ZB - 1) / ZB, ZB, 0, stream>>>(deg, n);
    }
    degree_kernel<<<(N_EDGES + ZB - 1) / ZB, ZB, 0, stream>>>(dst, deg, N_EDGES);

    // ---- layer 1 ----
    scatter_add128<<<N_EDGES / 8, 256, 0, stream>>>(x, src, dst, msg, N_EDGES);
    sage_gemm_wmma<HID, true><<<N_NODES / 16, 256, 0, stream>>>(
        x, msg, deg, W_self1, W_neigh1, b1, h1);

    // ---- layer 2 ----
    {
        int n = N_NODES * HID;
        zero_f32<<<(n + ZB - 1) / ZB, ZB, 0, stream>>>(msg, n);
    }
    scatter_add128<<<N_EDGES / 8, 256, 0, stream>>>(h1, src, dst, msg, N_EDGES);
    sage_gemm_wmma<OUT_DIM, false><<<N_NODES / 16, 128, 0, stream>>>(
        h1, msg, deg, W_self2, W_neigh2, b2, outp);
}